// SimGCF_90984587198549
// MI455X (gfx1250) — compile-verified
//
#include <hip/hip_runtime.h>
#include <hip/hip_bf16.h>

typedef __attribute__((ext_vector_type(2))) float v2f;
typedef __attribute__((ext_vector_type(8))) float v8f;

#define TPB 256

// ---------------------------------------------------------------------------
// init: emb0 = concat(user_emb, item_emb); acc = emb0; x_cur = emb0; x_next = 0
// ---------------------------------------------------------------------------
__global__ void simgcf_init(const float* __restrict__ ue,
                            const float* __restrict__ ie,
                            float* __restrict__ emb0_out,
                            float* __restrict__ acc,
                            float* __restrict__ xcur,
                            float* __restrict__ xnext,
                            int und, int total) {
    int i = blockIdx.x * TPB + threadIdx.x;
    if (i >= total) return;
    float v = (i < und) ? ue[i] : ie[i - und];
    emb0_out[i] = v;
    acc[i]      = v;
    xcur[i]     = v;
    xnext[i]    = 0.0f;
}

// ---------------------------------------------------------------------------
// SpMM scatter: one thread per (edge, dim). row/col/vals loads are uniform
// within each wave (e = gid>>6 is constant across 32 lanes) -> broadcast from
// L2. Gather x_cur[col*64+d] is coalesced in d. Hardware f32 atomic add into
// the L2-resident x_next (38.4 MB << 192 MB L2).
// ---------------------------------------------------------------------------
__global__ void simgcf_scatter(const int* __restrict__ row,
                               const int* __restrict__ col,
                               const float* __restrict__ vals,
                               const float* __restrict__ xcur,
                               float* __restrict__ xnext,
                               int nnz) {
    int gid = blockIdx.x * TPB + threadIdx.x;   // < 4.8M*64 = 307.2M, fits int32
    int e = gid >> 6;
    if (e >= nnz) return;
    int d = gid & 63;
    int r = row[e];
    int c = col[e];
    float v = vals[e];
    float x = xcur[(long)c * 64 + d];
    __hip_atomic_fetch_add(&xnext[(long)r * 64 + d], v * x,
                           __ATOMIC_RELAXED, __HIP_MEMORY_SCOPE_AGENT);
}

// ---------------------------------------------------------------------------
// acc += x_new ; zero the buffer that becomes next layer's x_next
// ---------------------------------------------------------------------------
__global__ void simgcf_accum(float* __restrict__ acc,
                             const float* __restrict__ xnew,
                             float* __restrict__ zero_buf,
                             int total) {
    int i = blockIdx.x * TPB + threadIdx.x;
    if (i >= total) return;
    acc[i] += xnew[i];
    zero_buf[i] = 0.0f;
}

// ---------------------------------------------------------------------------
// Finalize on the matrix pipe: out = 0.25 * (acc + x_last), computed per
// 16(row)x16(col) tile as a chain of 4x V_WMMA_F32_16X16X4_F32:
//   D = A_k x B_k + C,  A_k = S[:, 4k:4k+4] (data),  B_k = 0.25 * I-chunk.
// Identity matmul => each output element is exactly one product -> bit-exact
// 0.25 scale, no reduction-order error.
//
// Documented A (16x4 f32) layout: lane L: M = L%16 ; VGPR v holds K = v + 2*(L/16).
// B (4x16) assumed mirrored:      lane L: N = L%16 ; VGPR v holds K = v + 2*(L/16).
// C/D (16x16 f32) layout:         lane L: N = L%16 ; VGPR v holds M = v + 8*(L/16).
// One wave per tile; the guard is wave-uniform so EXEC is all-ones at WMMA.
// ---------------------------------------------------------------------------
__global__ void simgcf_finalize_wmma(const float* __restrict__ acc,
                                     const float* __restrict__ xl,
                                     float* __restrict__ out,
                                     int n_tiles) {
    int wave = (blockIdx.x * TPB + threadIdx.x) >> 5;
    int lane = threadIdx.x & 31;
    if (wave >= n_tiles) return;          // uniform across the wave

    int row_tile = wave >> 2;             // 16-row group of the [N,64] matrix
    int col_tile = wave & 3;              // 16-col group (64/16 = 4)
    int base_row = row_tile << 4;
    int base_col = col_tile << 4;

    int m  = lane & 15;
    int hi = lane >> 4;                   // 0: lanes 0-15, 1: lanes 16-31

    v8f c = {};
    #pragma unroll
    for (int k = 0; k < 4; ++k) {
        // A chunk (data): A[m, kk] = S[base_row+m, base_col + 4k + kk]
        int kk0 = 4 * k + 0 + 2 * hi;     // VGPR0 holds K = 0 + 2*hi
        int kk1 = 4 * k + 1 + 2 * hi;     // VGPR1 holds K = 1 + 2*hi
        long i0 = (long)(base_row + m) * 64 + base_col + kk0;
        long i1 = (long)(base_row + m) * 64 + base_col + kk1;
        v2f a;
        a.x = acc[i0] + xl[i0];
        a.y = acc[i1] + xl[i1];
        // B chunk (0.25 * identity): B[kk, n] = 0.25 iff n == 4k+kk (in tile)
        v2f b;
        b.x = (m == kk0) ? 0.25f : 0.0f;
        b.y = (m == kk1) ? 0.25f : 0.0f;
        c = __builtin_amdgcn_wmma_f32_16x16x4_f32(
                /*neg_a=*/false, a, /*neg_b=*/false, b,
                /*c_mod=*/(short)0, c, /*reuse_a=*/false, /*reuse_b=*/false);
    }

    // Store D per C/D layout
    int ocol = base_col + m;
    #pragma unroll
    for (int v = 0; v < 8; ++v) {
        int orow = base_row + v + 8 * hi;
        out[(long)orow * 64 + ocol] = c[v];
    }
}

// ---------------------------------------------------------------------------
extern "C" void kernel_launch(void* const* d_in, const int* in_sizes, int n_in,
                              void* d_out, int out_size, void* d_ws, size_t ws_size,
                              hipStream_t stream) {
    const float* user_emb = (const float*)d_in[0];
    const float* item_emb = (const float*)d_in[1];
    const int*   row      = (const int*)  d_in[2];
    const int*   col      = (const int*)  d_in[3];
    const float* vals     = (const float*)d_in[4];

    const int und   = in_sizes[0];              // U_N * 64
    const int total = in_sizes[0] + in_sizes[1];// N * 64 = 9,600,000
    const int nnz   = in_sizes[2];              // 4,800,000

    float* emb0_out = (float*)d_out;            // first output: emb0 [N*64]
    float* acc      = emb0_out + total;         // second output slot doubles as acc
    float* ws0      = (float*)d_ws;             // ping
    float* ws1      = ws0 + total;              // pong   (2 * 38.4 MB scratch)

    const int blocksN = (total + TPB - 1) / TPB;            // 37,500
    const long long ework = (long long)nnz * 64;            // 307.2M threads
    const int blocksE = (int)((ework + TPB - 1) / TPB);     // 1,200,000

    // init: emb0 / acc / x0 = concat ; x_next = 0
    simgcf_init<<<blocksN, TPB, 0, stream>>>(user_emb, item_emb, emb0_out,
                                             acc, ws0, ws1, und, total);

    // layer 1: x1 = A x0   (ws0 -> ws1)
    simgcf_scatter<<<blocksE, TPB, 0, stream>>>(row, col, vals, ws0, ws1, nnz);
    simgcf_accum<<<blocksN, TPB, 0, stream>>>(acc, ws1, ws0, total); // acc+=x1, zero ws0

    // layer 2: x2 = A x1   (ws1 -> ws0)
    simgcf_scatter<<<blocksE, TPB, 0, stream>>>(row, col, vals, ws1, ws0, nnz);
    simgcf_accum<<<blocksN, TPB, 0, stream>>>(acc, ws0, ws1, total); // acc+=x2, zero ws1

    // layer 3: x3 = A x2   (ws0 -> ws1)
    simgcf_scatter<<<blocksE, TPB, 0, stream>>>(row, col, vals, ws0, ws1, nnz);

    // out = 0.25 * (acc + x3) on the WMMA pipe, one wave per 16x16 tile
    const int n_tiles = total / 256;                        // 37,500 tiles
    const int blocksF = (n_tiles + 7) / 8;                  // 8 waves / block
    simgcf_finalize_wmma<<<blocksF, TPB, 0, stream>>>(acc, ws1, acc, n_tiles);
}